// Router_45956150067879
// MI455X (gfx1250) — compile-verified
//
#include <hip/hip_runtime.h>
#include <hip/hip_bf16.h>

// MoE router for MI455X (gfx1250, wave32).
// Output layout in d_out (concatenated flat, 4-byte elems):
//   [0 .. E*T)                     router_scores  (float)  [E,T]
//   [E*T .. E*T + E*T*H)           router_indices (int32)  [E*T, H]
//   [E*T + E*T*H .. +E*T)          router_probs   (float)  [E*T, 1]

#define T_TOKENS 4096
#define HIDDEN   2048
#define N_EXPERTS 8

typedef __attribute__((ext_vector_type(2))) float v2f;
typedef __attribute__((ext_vector_type(8))) float v8f;
typedef __attribute__((ext_vector_type(4))) int   v4i;

// ---------------------------------------------------------------------------
// Kernel 1: logits GEMM (WMMA fp32 16x16x4) + top-2 + sigmoid.
// 256 threads = 8 waves; each wave owns 16 tokens; block covers 128 tokens.
// B is 4x16 with N=0..7 the real experts; lanes 8..15 alias rows 0..7 so the
// load stream is branch-free.  C columns 8..15 contain duplicates and are
// simply never read (each D[m][n] depends only on B column n).
// ---------------------------------------------------------------------------
__global__ __launch_bounds__(256) void router_topk_kernel(
    const float* __restrict__ hs,      // [T, H]
    const float* __restrict__ wr,      // [E, H]
    float* __restrict__ scores,        // [E*T]
    float* __restrict__ probs)         // [E*T]
{
    // per-wave 16x16 logits tile staging (pad to 17 to dodge bank conflicts)
    __shared__ float lds_logits[8][16][17];

    const int wave = threadIdx.x >> 5;       // 0..7
    const int lane = threadIdx.x & 31;       // wave32
    const int half = lane >> 4;              // 0: K=0,1  1: K=2,3  (32-bit A layout)
    const int l16  = lane & 15;
    const int tokBase = blockIdx.x * 128 + wave * 16;

    // A lane base: row M = l16 (token), K pair selected by half.
    const float* aBase = hs + (size_t)(tokBase + l16) * HIDDEN + 2 * half;
    // B lane base: column N = l16 (expert); lanes 8..15 duplicate rows 0..7.
    const float* bBase = wr + (size_t)(l16 & (N_EXPERTS - 1)) * HIDDEN + 2 * half;

    v8f c = {};
    for (int k0 = 0; k0 < HIDDEN; k0 += 4) {
        v2f a = *(const v2f*)(aBase + k0);           // K = k0+2h, k0+2h+1
        v2f b = *(const v2f*)(bBase + k0);
        // D = A(16x4,f32) * B(4x16,f32) + C(16x16,f32)
        c = __builtin_amdgcn_wmma_f32_16x16x4_f32(
                /*neg_a=*/false, a, /*neg_b=*/false, b,
                /*c_mod=*/(short)0, c, /*reuse_a=*/false, /*reuse_b=*/false);
    }

    // C layout: VGPR j -> M = j + 8*half (token-in-tile), N = l16 (expert col)
    #pragma unroll
    for (int j = 0; j < 8; ++j) {
        lds_logits[wave][j + 8 * half][l16] = c[j];
    }
    __syncthreads();

    // Lanes 0..15 of each wave: one token each -> top-2 over 8 experts, sigmoid,
    // scatter into dense [E,T] grid (non-top entries are sigmoid(-inf) == 0).
    if (lane < 16) {
        const int tok = tokBase + lane;
        float v[N_EXPERTS];
        #pragma unroll
        for (int e = 0; e < N_EXPERTS; ++e) v[e] = lds_logits[wave][lane][e];

        // top-1 (ties -> lowest index, matching jax.lax.top_k)
        int i1 = 0; float v1 = v[0];
        #pragma unroll
        for (int e = 1; e < N_EXPERTS; ++e) { if (v[e] > v1) { v1 = v[e]; i1 = e; } }
        // top-2
        int i2 = -1; float v2 = -__builtin_inff();
        #pragma unroll
        for (int e = 0; e < N_EXPERTS; ++e) {
            if (e != i1 && v[e] > v2) { v2 = v[e]; i2 = e; }
        }

        #pragma unroll
        for (int e = 0; e < N_EXPERTS; ++e) {
            float s = 0.0f;
            if (e == i1) s = 1.0f / (1.0f + __expf(-v1));
            else if (e == i2) s = 1.0f / (1.0f + __expf(-v2));
            scores[(size_t)e * T_TOKENS + tok] = s;
            probs [(size_t)e * T_TOKENS + tok] = s;   // probs == scores.reshape(-1,1)
        }
    }
}

// ---------------------------------------------------------------------------
// Kernel 2: router_indices fill -- 268 MB of broadcast stores, the real cost
// (~11.5 us at 23.3 TB/s HBM).  One block per (e,t) row; 512 int4 per row;
// 256 threads x 2 int4 each.  Non-temporal b128 stores: write-only stream,
// keep it out of WGP$/L2.
// ---------------------------------------------------------------------------
__global__ __launch_bounds__(256) void router_indices_fill(int* __restrict__ idx)
{
    const int row = blockIdx.x;                 // 0 .. E*T-1
    const int t = row & (T_TOKENS - 1);         // value broadcast across the row
    v4i v;
    v.x = t; v.y = t; v.z = t; v.w = t;
    v4i* p = (v4i*)idx + (size_t)row * (HIDDEN / 4);
    __builtin_nontemporal_store(v, p + threadIdx.x);
    __builtin_nontemporal_store(v, p + 256 + threadIdx.x);
}

extern "C" void kernel_launch(void* const* d_in, const int* in_sizes, int n_in,
                              void* d_out, int out_size, void* d_ws, size_t ws_size,
                              hipStream_t stream) {
    const float* hs = (const float*)d_in[0];   // [T, H] fp32
    const float* wr = (const float*)d_in[1];   // [E, H] fp32
    // d_in[2] = top_k scalar (compile-time 2 here)

    const size_t ET = (size_t)N_EXPERTS * T_TOKENS;
    float* out     = (float*)d_out;
    float* scores  = out;                            // [E,T]
    int*   indices = (int*)(out + ET);               // [E*T, H]
    float* probs   = out + ET + ET * (size_t)HIDDEN; // [E*T, 1]

    router_topk_kernel<<<T_TOKENS / 128, 256, 0, stream>>>(hs, wr, scores, probs);
    router_indices_fill<<<(int)ET, 256, 0, stream>>>(indices);
}